// Learnable_Frequency_Band_MLP_29935922053590
// MI455X (gfx1250) — compile-verified
//
#include <hip/hip_runtime.h>
#include <hip/hip_bf16.h>
#include <math.h>

// ---- problem constants (from reference) ----
#define BATCH   32
#define CCH     24
#define TLEN    1024
#define DDIM    128
#define EDIM    128
#define NBANDS  3
#define NFREQ   513            // T/2 + 1
#define BC      (BATCH*CCH)    // 768
#define ROWS    (BC*NBANDS)    // 2304  (GEMM M)
#define KDIM    256            // GEMM K = 2*D
#define NDIM    256            // GEMM N = 2*E
#define LAM     0.01f
#define SCALE   0.03125f       // 1/sqrt(1024)
#define TWOPI   6.2831853071795864769f

typedef __attribute__((ext_vector_type(2))) float v2f;
typedef __attribute__((ext_vector_type(8))) float v8f;

// ---- workspace layout (bytes) ----
#define WS_K1   0                              // 6*1024 f32            = 24576
#define WS_XB   24576                          // BC*6*128 f32          = 2359296
#define WS_AM   (WS_XB + 2359296)              // 2304*256 f32          = 2359296
#define WS_BM   (WS_AM + 2359296)              // 256*256 f32           = 262144
#define WS_BIAS (WS_BM + 262144)               // 256 f32               = 1024
#define WS_Y    (WS_BIAS + 1024)               // BC*6*128 f32          = 2359296

// =====================================================================
// Kernel 1: K[j][t], j=2n -> Re, j=2n+1 -> Im of band-summed DFT rows.
// K[2n][t]   =  bw[n]*(1/32)*sum_{f in band n} cos(2*pi*f*t/T)
// K[2n+1][t] = -bw[n]*(1/32)*sum_{f in band n} sin(2*pi*f*t/T)
// =====================================================================
__global__ void compute_k1_kernel(const float* __restrict__ bb,
                                  const float* __restrict__ bw,
                                  float* __restrict__ K1) {
    int t = blockIdx.x * blockDim.x + threadIdx.x;   // 0..1023
    if (t >= TLEN) return;
    float edges[NBANDS + 1];
    edges[0] = 0.0f; edges[1] = bb[0]; edges[2] = bb[1]; edges[3] = 1.0f;
    int idx[NBANDS + 1];
    #pragma unroll
    for (int k = 0; k <= NBANDS; ++k)
        idx[k] = (int)floorf(edges[k] * (float)(NFREQ - 1));
    for (int n = 0; n < NBANDS; ++n) {
        float ac = 0.0f, as = 0.0f;
        for (int f = idx[n]; f < idx[n + 1]; ++f) {
            int ph = (f * t) & (TLEN - 1);           // exact mod-T phase
            float th = (TWOPI / (float)TLEN) * (float)ph;
            float s, c; __sincosf(th, &s, &c);
            ac += c; as += s;
        }
        float w = bw[n] * SCALE;
        K1[(2 * n) * TLEN + t]     =  w * ac;
        K1[(2 * n + 1) * TLEN + t] = -w * as;
    }
}

// =====================================================================
// Kernel 2: xb[bc][j][d] = sum_t K[j][t] * x[bc][t][d]   (j=0..5)
// One block per (b,c). 256 thr = 8 t-groups x 32 d-threads(float4).
// Streams 512 KB of x per block with b128 loads; K in LDS.
// =====================================================================
__global__ void stage1_kernel(const float* __restrict__ x,
                              const float* __restrict__ K1,
                              float* __restrict__ xb) {
    __shared__ float lds[6 * TLEN];                  // 24 KB, reused for reduce
    const int bc  = blockIdx.x;
    const int tid = threadIdx.x;
    for (int i = tid; i < 6 * TLEN; i += 256) lds[i] = K1[i];
    __syncthreads();

    const int dt = tid & 31;          // float4 column group: d = 4*dt..4*dt+3
    const int tg = tid >> 5;          // t-group 0..7
    const float4* xr = (const float4*)(x + (size_t)bc * TLEN * DDIM);

    float4 acc[6];
    #pragma unroll
    for (int j = 0; j < 6; ++j) acc[j] = make_float4(0.f, 0.f, 0.f, 0.f);

    for (int t = tg; t < TLEN; t += 8) {
        float4 v = xr[t * 32 + dt];
        #pragma unroll
        for (int j = 0; j < 6; ++j) {
            float k = lds[j * TLEN + t];
            acc[j].x = fmaf(k, v.x, acc[j].x);
            acc[j].y = fmaf(k, v.y, acc[j].y);
            acc[j].z = fmaf(k, v.z, acc[j].z);
            acc[j].w = fmaf(k, v.w, acc[j].w);
        }
    }
    __syncthreads();                                  // done reading K
    float4* p = (float4*)lds;                         // 8*6*32 = 1536 float4
    #pragma unroll
    for (int j = 0; j < 6; ++j) p[(tg * 6 + j) * 32 + dt] = acc[j];
    __syncthreads();

    if (tid < 192) {                                  // (j,dt) pairs: 6*32
        int j = tid >> 5, d4 = tid & 31;
        float4 s = p[(0 * 6 + j) * 32 + d4];
        #pragma unroll
        for (int g = 1; g < 8; ++g) {
            float4 q = p[(g * 6 + j) * 32 + d4];
            s.x += q.x; s.y += q.y; s.z += q.z; s.w += q.w;
        }
        ((float4*)(xb + ((size_t)bc * 6 + j) * DDIM))[d4] = s;
    }
}

// =====================================================================
// Kernel 3a: A[row][col], row = bc*3+n.
//   col<128 : xr*Wr - xi*Wi   |  col>=128 : xr*Wi + xi*Wr   (d = col%128)
// =====================================================================
__global__ void build_a_kernel(const float* __restrict__ xb,
                               const float* __restrict__ cw,   // [NB,C,D,2]
                               float* __restrict__ A) {
    int g = blockIdx.x * blockDim.x + threadIdx.x;   // 0..2304*256-1
    int row = g >> 8, col = g & 255;
    int bc = row / NBANDS, n = row % NBANDS;
    int c  = bc % CCH;
    int d  = col & 127;
    float xr = xb[((size_t)bc * 6 + 2 * n) * DDIM + d];
    float xi = xb[((size_t)bc * 6 + 2 * n + 1) * DDIM + d];
    const float* wp = cw + (((size_t)n * CCH + c) * DDIM + d) * 2;
    float Wr = wp[0], Wi = wp[1];
    A[g] = (col < 128) ? fmaf(xr, Wr, -xi * Wi) : fmaf(xr, Wi, xi * Wr);
}

// =====================================================================
// Kernel 3b: Bm = [[r, iw],[-iw, r]] (256x256), bias = [rb | ib]
// =====================================================================
__global__ void build_b_kernel(const float* __restrict__ r,
                               const float* __restrict__ iw,
                               const float* __restrict__ rb,
                               const float* __restrict__ ib,
                               float* __restrict__ Bm,
                               float* __restrict__ bias) {
    int g = blockIdx.x * blockDim.x + threadIdx.x;   // 0..65535
    int k = g >> 8, e = g & 255;
    float v;
    if (k < 128) v = (e < 128) ? r[k * EDIM + e] : iw[k * EDIM + (e - 128)];
    else         v = (e < 128) ? -iw[(k - 128) * EDIM + e] : r[(k - 128) * EDIM + (e - 128)];
    Bm[g] = v;
    if (g < 256) bias[g] = (g < 128) ? rb[g] : ib[g - 128];
}

// =====================================================================
// Kernel 4: WMMA GEMM  O(2304x256) = A(2304x256) x Bm(256x256) + bias,
// fused relu + softshrink, scattered into Y[bc][j][e] for the irfft.
// One wave per 16x16 tile; 64x v_wmma_f32_16x16x4_f32 K-loop.
// A/Bm are L2-resident (2.6 MB total) -> per-lane gathers are cheap.
// =====================================================================
__global__ void gemm_wmma_kernel(const float* __restrict__ A,
                                 const float* __restrict__ Bm,
                                 const float* __restrict__ bias,
                                 float* __restrict__ Y) {
    const int lane = threadIdx.x & 31;
    const int wave = threadIdx.x >> 5;
    const int tile = blockIdx.x * 8 + wave;          // 0..2303
    const int tm = tile >> 4;                        // 0..143 row tile
    const int tn = tile & 15;                        // 0..15  col tile

    const int mcol = lane & 15;
    const int koff = (lane >> 4) << 1;               // lanes 0-15:K+0,1; 16-31:K+2,3
    const int rowA = tm * 16 + mcol;                 // A row for this lane
    const int col  = tn * 16 + mcol;                 // output / Bm column

    v8f c;
    float bv = bias[col];
    #pragma unroll
    for (int i = 0; i < 8; ++i) c[i] = bv;

    #pragma unroll 4
    for (int k0 = 0; k0 < KDIM; k0 += 4) {
        v2f a, b;
        a[0] = A[(size_t)rowA * KDIM + k0 + koff];
        a[1] = A[(size_t)rowA * KDIM + k0 + koff + 1];
        b[0] = Bm[(size_t)(k0 + koff) * NDIM + col];
        b[1] = Bm[(size_t)(k0 + koff + 1) * NDIM + col];
        c = __builtin_amdgcn_wmma_f32_16x16x4_f32(
                /*neg_a=*/false, a, /*neg_b=*/false, b,
                /*c_mod=*/(short)0, c, /*reuse_a=*/false, /*reuse_b=*/false);
    }

    const int half = col >> 7;                       // 0 -> Re, 1 -> Im
    const int e    = col & 127;
    #pragma unroll
    for (int i = 0; i < 8; ++i) {
        int m   = i + ((lane >> 4) << 3);            // C/D layout: lanes>=16 -> M+8
        int row = tm * 16 + m;                       // = bc*3 + n
        int bc  = row / NBANDS, n = row % NBANDS;
        float v = c[i];
        v = fmaxf(v, 0.0f);                          // relu
        v = (v > LAM) ? (v - LAM) : 0.0f;            // softshrink (v>=0 already)
        Y[((size_t)bc * 6 + 2 * n + half) * EDIM + e] = v;
    }
}

// =====================================================================
// Kernel 5: irfft expansion (3 bins -> T=1024), writes 402 MB.
// out[t][e] = (1/32)*( Y0r + 2*(c1*Y1r - s1*Y1i) + 2*(c2*Y2r - s2*Y2i) )
// grid (768, 4): block = one (b,c) x 256-t chunk; Y row in LDS; f4 stores.
// =====================================================================
__global__ void irfft_kernel(const float* __restrict__ Yw,
                             float* __restrict__ out) {
    __shared__ float Y[6 * EDIM];                    // 3 KB
    const int bc = blockIdx.x;
    const int tid = threadIdx.x;
    for (int i = tid; i < 6 * EDIM; i += 256) Y[i] = Yw[(size_t)bc * 6 * EDIM + i];
    __syncthreads();

    const int dt = tid & 31;                         // e float4 group
    const int tg = tid >> 5;                         // t subgroup 0..7
    const float4* Y4 = (const float4*)Y;
    const float4 y0 = Y4[0 * 32 + dt];
    const float4 a1 = Y4[2 * 32 + dt];               // Y1.re
    const float4 b1 = Y4[3 * 32 + dt];               // Y1.im
    const float4 a2 = Y4[4 * 32 + dt];               // Y2.re
    const float4 b2 = Y4[5 * 32 + dt];               // Y2.im

    const int tbase = blockIdx.y * 256;
    float4* op = (float4*)(out + (size_t)bc * TLEN * EDIM);

    for (int t = tbase + tg; t < tbase + 256; t += 8) {
        float th = (TWOPI / (float)TLEN) * (float)t;
        float s1, c1; __sincosf(th, &s1, &c1);
        float c2 = 1.0f - 2.0f * s1 * s1;
        float s2 = 2.0f * s1 * c1;
        float4 o;
        o.x = SCALE * (y0.x + 2.0f * (c1 * a1.x - s1 * b1.x + c2 * a2.x - s2 * b2.x));
        o.y = SCALE * (y0.y + 2.0f * (c1 * a1.y - s1 * b1.y + c2 * a2.y - s2 * b2.y));
        o.z = SCALE * (y0.z + 2.0f * (c1 * a1.z - s1 * b1.z + c2 * a2.z - s2 * b2.z));
        o.w = SCALE * (y0.w + 2.0f * (c1 * a1.w - s1 * b1.w + c2 * a2.w - s2 * b2.w));
        op[t * 32 + dt] = o;
    }
}

// =====================================================================
extern "C" void kernel_launch(void* const* d_in, const int* in_sizes, int n_in,
                              void* d_out, int out_size, void* d_ws, size_t ws_size,
                              hipStream_t stream) {
    const float* x  = (const float*)d_in[0];
    const float* bb = (const float*)d_in[1];
    const float* bw = (const float*)d_in[2];
    const float* cw = (const float*)d_in[3];
    const float* r  = (const float*)d_in[4];
    const float* iw = (const float*)d_in[5];
    const float* rb = (const float*)d_in[6];
    const float* ib = (const float*)d_in[7];
    float* out = (float*)d_out;

    char* ws = (char*)d_ws;
    float* K1   = (float*)(ws + WS_K1);
    float* XB   = (float*)(ws + WS_XB);
    float* AM   = (float*)(ws + WS_AM);
    float* BM   = (float*)(ws + WS_BM);
    float* BIAS = (float*)(ws + WS_BIAS);
    float* YW   = (float*)(ws + WS_Y);

    compute_k1_kernel<<<4, 256, 0, stream>>>(bb, bw, K1);
    stage1_kernel<<<BC, 256, 0, stream>>>(x, K1, XB);
    build_a_kernel<<<(ROWS * KDIM) / 256, 256, 0, stream>>>(XB, cw, AM);
    build_b_kernel<<<(KDIM * NDIM) / 256, 256, 0, stream>>>(r, iw, rb, ib, BM, BIAS);
    gemm_wmma_kernel<<<ROWS / 8, 256, 0, stream>>>(AM, BM, BIAS, YW);
    irfft_kernel<<<dim3(BC, 4), 256, 0, stream>>>(YW, out);
}